// Attention_48962627175197
// MI455X (gfx1250) — compile-verified
//
#include <hip/hip_runtime.h>
#include <hip/hip_bf16.h>

typedef __attribute__((ext_vector_type(2))) float v2f;
typedef __attribute__((ext_vector_type(8))) float v8f;

#define INTER 128
#define N_SP  1024   // 32*32 subsampled spatial
#define C_IN  256
#define C_OUT 256

__device__ __forceinline__ v8f wmma_f32_k4(v2f a, v2f b, v8f c) {
  // D = A(16x4) * B(4x16) + C, fp32 exact
  return __builtin_amdgcn_wmma_f32_16x16x4_f32(false, a, false, b, (short)0, c,
                                               false, false);
}

__device__ __forceinline__ v8f zero8() {
  v8f z = {0.f, 0.f, 0.f, 0.f, 0.f, 0.f, 0.f, 0.f};
  return z;
}

// ---------------------------------------------------------------------------
// Kernel 1: strided subsample + Q/K/V projections.
// grid (64 n-tiles, 3 mats, 8 batch), block 32 (1 wave).
// Each wave: 8 M-tiles (rows 0..127 of Wq|Wk|Wv) x 16 spatial cols, K=256.
// ---------------------------------------------------------------------------
__global__ void qkv_kernel(const float* __restrict__ x,
                           const float* __restrict__ Wq,
                           const float* __restrict__ Wk,
                           const float* __restrict__ Wv,
                           float* __restrict__ qkv) {
  const int lane = threadIdx.x;
  const int half = lane >> 4;
  const int lid  = lane & 15;
  const int n0   = blockIdx.x * 16;
  const int mat  = blockIdx.y;
  const int b    = blockIdx.z;

  const float* W = (mat == 0) ? Wq : (mat == 1) ? Wk : Wv;

  const int n  = n0 + lid;
  const int hs = n >> 5;
  const int ws = n & 31;
  // x[b][c][hs*4][ws*4]; channel stride = 128*128 = 16384 floats
  const float* xb = x + (size_t)b * C_IN * 16384 + (size_t)(hs * 4) * 128 + ws * 4;

  v8f acc[8];
  #pragma unroll
  for (int mt = 0; mt < 8; ++mt) acc[mt] = zero8();

  for (int kk = 0; kk < C_IN; kk += 4) {
    const int c0 = kk + half * 2;
    v2f bf;
    bf.x = xb[(size_t)c0 * 16384];
    bf.y = xb[(size_t)(c0 + 1) * 16384];
    #pragma unroll
    for (int mt = 0; mt < 8; ++mt) {
      const int row = mt * 16 + lid;
      v2f af;
      af.x = W[row * C_IN + c0];
      af.y = W[row * C_IN + c0 + 1];
      acc[mt] = wmma_f32_k4(af, bf, acc[mt]);
    }
  }

  float* dst = qkv + (size_t)(b * 3 + mat) * INTER * N_SP;
  #pragma unroll
  for (int mt = 0; mt < 8; ++mt) {
    #pragma unroll
    for (int r = 0; r < 8; ++r) {
      const int row = mt * 16 + half * 8 + r;
      dst[(size_t)row * N_SP + n0 + lid] = acc[mt][r];
    }
  }
}

// ---------------------------------------------------------------------------
// Kernel 2: scores = q^T k (K=128) fused with row softmax -> gate.
// grid (64 n-tiles, 8 batch), block 128 (4 waves).
// Each wave keeps a 16x256 score stripe in 128 accumulator VGPRs; softmax
// stats via shfl_xor within the wave + tiny LDS exchange across waves.
// ---------------------------------------------------------------------------
__global__ void attn_kernel(const float* __restrict__ qkv,
                            float* __restrict__ gate) {
  const int tid  = threadIdx.x;
  const int w    = tid >> 5;
  const int lane = tid & 31;
  const int half = lane >> 4;
  const int lid  = lane & 15;
  const int n0   = blockIdx.x * 16;
  const int b    = blockIdx.y;

  const float* q = qkv + (size_t)(b * 3 + 0) * INTER * N_SP;
  const float* k = qkv + (size_t)(b * 3 + 1) * INTER * N_SP;

  v8f acc[16];
  #pragma unroll
  for (int j = 0; j < 16; ++j) acc[j] = zero8();

  for (int kk = 0; kk < INTER; kk += 4) {
    const int c0 = kk + half * 2;
    v2f af;
    af.x = q[(size_t)c0 * N_SP + n0 + lid];
    af.y = q[(size_t)(c0 + 1) * N_SP + n0 + lid];
    #pragma unroll
    for (int j = 0; j < 16; ++j) {
      const int m0 = (w * 16 + j) * 16;
      v2f bf;
      bf.x = k[(size_t)c0 * N_SP + m0 + lid];
      bf.y = k[(size_t)(c0 + 1) * N_SP + m0 + lid];
      acc[j] = wmma_f32_k4(af, bf, acc[j]);
    }
  }

  // This lane holds rows (half*8 + r), r=0..7, for columns m0+lid.
  float rmax[8];
  #pragma unroll
  for (int r = 0; r < 8; ++r) rmax[r] = -3.402823466e38f;
  #pragma unroll
  for (int j = 0; j < 16; ++j) {
    #pragma unroll
    for (int r = 0; r < 8; ++r) rmax[r] = fmaxf(rmax[r], acc[j][r]);
  }
  #pragma unroll
  for (int m = 1; m < 16; m <<= 1) {
    #pragma unroll
    for (int r = 0; r < 8; ++r)
      rmax[r] = fmaxf(rmax[r], __shfl_xor(rmax[r], m, 32));
  }

  __shared__ float smax[4][16];
  __shared__ float ssum[4][16];
  if (lid == 0) {
    #pragma unroll
    for (int r = 0; r < 8; ++r) smax[w][half * 8 + r] = rmax[r];
  }
  __syncthreads();

  float gmax[8];
  #pragma unroll
  for (int r = 0; r < 8; ++r) {
    const int row = half * 8 + r;
    float gm = fmaxf(fmaxf(smax[0][row], smax[1][row]),
                     fmaxf(smax[2][row], smax[3][row]));
    gmax[r] = gm;
  }

  float rsum[8];
  #pragma unroll
  for (int r = 0; r < 8; ++r) rsum[r] = 0.f;
  #pragma unroll
  for (int j = 0; j < 16; ++j) {
    #pragma unroll
    for (int r = 0; r < 8; ++r) {
      float e = __expf(acc[j][r] - gmax[r]);
      acc[j][r] = e;
      rsum[r] += e;
    }
  }
  #pragma unroll
  for (int m = 1; m < 16; m <<= 1) {
    #pragma unroll
    for (int r = 0; r < 8; ++r) rsum[r] += __shfl_xor(rsum[r], m, 32);
  }
  if (lid == 0) {
    #pragma unroll
    for (int r = 0; r < 8; ++r) ssum[w][half * 8 + r] = rsum[r];
  }
  __syncthreads();

  float ginv[8];
  #pragma unroll
  for (int r = 0; r < 8; ++r) {
    const int row = half * 8 + r;
    ginv[r] = 1.f / (ssum[0][row] + ssum[1][row] + ssum[2][row] + ssum[3][row]);
  }

  float* gb = gate + (size_t)b * N_SP * N_SP;
  #pragma unroll
  for (int j = 0; j < 16; ++j) {
    const int m0 = (w * 16 + j) * 16;
    #pragma unroll
    for (int r = 0; r < 8; ++r) {
      const int row = n0 + half * 8 + r;
      gb[(size_t)row * N_SP + m0 + lid] = acc[j][r] * ginv[r];
    }
  }
}

// ---------------------------------------------------------------------------
// Kernel 3: gv[c][n] = sum_m v[c][m] * gate[n][m]   (K = 1024)
// grid (64 n-tiles, 8 c-tiles, 8 batch), block 32.
// Both operands contiguous along m -> packed v2f (b64) loads.
// ---------------------------------------------------------------------------
__global__ void gv_kernel(const float* __restrict__ qkv,
                          const float* __restrict__ gate,
                          float* __restrict__ gv) {
  const int lane = threadIdx.x;
  const int half = lane >> 4;
  const int lid  = lane & 15;
  const int n0   = blockIdx.x * 16;
  const int c0   = blockIdx.y * 16;
  const int b    = blockIdx.z;

  const float* v  = qkv + (size_t)(b * 3 + 2) * INTER * N_SP;
  const float* gb = gate + (size_t)b * N_SP * N_SP;

  v8f acc = zero8();
  #pragma unroll 4
  for (int mm = 0; mm < N_SP; mm += 4) {
    const int m0 = mm + half * 2;
    v2f af = *(const v2f*)(v + (size_t)(c0 + lid) * N_SP + m0);
    v2f bf = *(const v2f*)(gb + (size_t)(n0 + lid) * N_SP + m0);
    acc = wmma_f32_k4(af, bf, acc);
  }

  #pragma unroll
  for (int r = 0; r < 8; ++r) {
    const int row = c0 + half * 8 + r;
    gv[(size_t)(b * INTER + row) * N_SP + n0 + lid] = acc[r];
  }
}

// ---------------------------------------------------------------------------
// Kernel 4: channel attention: mean over spatial -> conv1d(3) -> sigmoid.
// grid (8), block 128 (one thread per channel).
// ---------------------------------------------------------------------------
__global__ void ca_kernel(const float* __restrict__ gv,
                          const float* __restrict__ w1d,
                          float* __restrict__ ca) {
  const int b = blockIdx.x;
  const int c = threadIdx.x;
  __shared__ float sm[INTER];

  const float* g = gv + (size_t)(b * INTER + c) * N_SP;
  float s = 0.f;
  for (int n = 0; n < N_SP; ++n) s += g[n];
  sm[c] = s * (1.0f / 1024.0f);
  __syncthreads();

  const float left  = (c > 0) ? sm[c - 1] : 0.f;
  const float mid   = sm[c];
  const float right = (c < INTER - 1) ? sm[c + 1] : 0.f;
  const float val   = w1d[0] * left + w1d[1] * mid + w1d[2] * right;
  ca[b * INTER + c] = 1.0f / (1.0f + __expf(-val));
}

// ---------------------------------------------------------------------------
// Kernel 5: 3x3 SAME conv 128->256 over 32x32 as implicit GEMM, K = 128*9.
// K packed as ki = i*9 + (dy*3+dx) so A (Wc) loads are contiguous.
// grid (64 n-tiles, 16 o-tiles, 8 batch), block 32.
// ---------------------------------------------------------------------------
__device__ __forceinline__ float conv_in(const float* __restrict__ gvb,
                                         const float* __restrict__ sca,
                                         int kz, int y, int x) {
  const int i  = kz / 9;
  const int t  = kz - i * 9;
  const int dy = t / 3;
  const int dx = t - dy * 3;
  const int yy = y + dy - 1;
  const int xx = x + dx - 1;
  if ((unsigned)yy < 32u && (unsigned)xx < 32u)
    return gvb[(size_t)i * N_SP + (yy << 5) + xx] * sca[i];
  return 0.f;
}

__global__ void conv_kernel(const float* __restrict__ gv,
                            const float* __restrict__ ca,
                            const float* __restrict__ Wc,
                            float* __restrict__ out) {
  const int lane = threadIdx.x;
  const int half = lane >> 4;
  const int lid  = lane & 15;
  const int n0   = blockIdx.x * 16;
  const int o0   = blockIdx.y * 16;
  const int b    = blockIdx.z;

  __shared__ float sca[INTER];
  #pragma unroll
  for (int t = 0; t < 4; ++t) sca[lane + 32 * t] = ca[b * INTER + lane + 32 * t];
  __syncthreads();

  const float* gvb = gv + (size_t)b * INTER * N_SP;
  const int n = n0 + lid;
  const int y = n >> 5;
  const int xq = n & 31;

  v8f acc = zero8();
  for (int kk = 0; kk < INTER * 9; kk += 4) {
    const int kb = kk + half * 2;
    v2f af;
    af.x = Wc[(size_t)(o0 + lid) * (INTER * 9) + kb];
    af.y = Wc[(size_t)(o0 + lid) * (INTER * 9) + kb + 1];
    v2f bf;
    bf.x = conv_in(gvb, sca, kb, y, xq);
    bf.y = conv_in(gvb, sca, kb + 1, y, xq);
    acc = wmma_f32_k4(af, bf, acc);
  }

  #pragma unroll
  for (int r = 0; r < 8; ++r) {
    const int o = o0 + half * 8 + r;
    out[(size_t)(b * C_OUT + o) * N_SP + n0 + lid] = acc[r];
  }
}

// ---------------------------------------------------------------------------
extern "C" void kernel_launch(void* const* d_in, const int* in_sizes, int n_in,
                              void* d_out, int out_size, void* d_ws, size_t ws_size,
                              hipStream_t stream) {
  const float* x   = (const float*)d_in[0];
  const float* Wq  = (const float*)d_in[1];
  const float* Wk  = (const float*)d_in[2];
  const float* Wv  = (const float*)d_in[3];
  const float* w1d = (const float*)d_in[4];
  const float* Wc  = (const float*)d_in[5];

  float* out  = (float*)d_out;                  // 8*256*32*32 = 2,097,152
  float* gate = out + (size_t)8 * 256 * 1024;   // 8*1024*1024 = 8,388,608
  float* ca   = gate + (size_t)8 * 1024 * 1024; // 8*128       = 1,024

  float* qkv = (float*)d_ws;                         // 8*3*128*1024 floats
  float* gv  = qkv + (size_t)8 * 3 * INTER * N_SP;   // 8*128*1024 floats

  qkv_kernel <<<dim3(64, 3, 8),  32,  0, stream>>>(x, Wq, Wk, Wv, qkv);
  attn_kernel<<<dim3(64, 8),     128, 0, stream>>>(qkv, gate);
  gv_kernel  <<<dim3(64, 8, 8),  32,  0, stream>>>(qkv, gate, gv);
  ca_kernel  <<<dim3(8),         128, 0, stream>>>(gv, w1d, ca);
  conv_kernel<<<dim3(64, 16, 8), 32,  0, stream>>>(gv, ca, Wc, out);
}